// VCSMC_69844758167651
// MI455X (gfx1250) — compile-verified
//
#include <hip/hip_runtime.h>
#include <hip/hip_bf16.h>
#include <stdint.h>

// Problem constants (match reference)
#define KP 256      // particles
#define TS 16       // live subtrees
#define RM 16       // prior merges
#define SS 1024     // sites
#define AD 4        // alphabet
#define DE 64       // embedding dim
#define CF 16       // site feature dim

typedef __attribute__((ext_vector_type(2))) float v2f;
typedef __attribute__((ext_vector_type(8))) float v8f;

// ---------------------------------------------------------------------------
// Threefry-2x32 (mirrors jax._src.prng.threefry2x32: 20 rounds, 5 injections)
// ---------------------------------------------------------------------------
__device__ __forceinline__ uint32_t rotl32(uint32_t x, int r) {
  return (x << r) | (x >> (32 - r));
}

__device__ void threefry2x32(uint32_t k0, uint32_t k1, uint32_t x0, uint32_t x1,
                             uint32_t& o0, uint32_t& o1) {
  uint32_t ks2 = k0 ^ k1 ^ 0x1BD11BDAu;
  uint32_t v0 = x0 + k0, v1 = x1 + k1;
#define TFR(r) { v0 += v1; v1 = rotl32(v1, (r)); v1 ^= v0; }
  TFR(13) TFR(15) TFR(26) TFR(6)   v0 += k1;  v1 += ks2 + 1u;
  TFR(17) TFR(29) TFR(16) TFR(24)  v0 += ks2; v1 += k0 + 2u;
  TFR(13) TFR(15) TFR(26) TFR(6)   v0 += k0;  v1 += k1 + 3u;
  TFR(17) TFR(29) TFR(16) TFR(24)  v0 += k1;  v1 += ks2 + 4u;
  TFR(13) TFR(15) TFR(26) TFR(6)   v0 += ks2; v1 += k0 + 5u;
#undef TFR
  o0 = v0; o1 = v1;
}

// element e of jax random_bits for an even total count n (iota split in halves)
__device__ __forceinline__ uint32_t rbits_even(uint32_t k0, uint32_t k1,
                                               uint32_t e, uint32_t n) {
  uint32_t half = n >> 1, o0, o1;
  if (e < half) { threefry2x32(k0, k1, e, e + half, o0, o1); return o0; }
  threefry2x32(k0, k1, e - half, e, o0, o1); return o1;
}

__device__ __forceinline__ float jax_uniform(uint32_t bits, float mn, float mx) {
  uint32_t u = (bits >> 9) | 0x3f800000u;
  float f = __uint_as_float(u) - 1.0f;
  return fmaxf(mn, f * (mx - mn) + mn);
}

// ---------------------------------------------------------------------------
// Kernel 1: PRNG + resampling + merge bookkeeping + scalar weight terms.
// One block per particle; one thread per categorical class; LDS argmax
// reduction with first-index tie-breaking (matches jnp.argmax).
// ---------------------------------------------------------------------------
__global__ __launch_bounds__(256) void k_setup(
    const float* __restrict__ logw, const float* __restrict__ b1r,
    const float* __restrict__ b2r, const int* __restrict__ lc,
    const float* __restrict__ logpi, const float* __restrict__ ldf,
    int* __restrict__ idxP, int* __restrict__ idx1a, int* __restrict__ idx2a,
    float* __restrict__ br1, float* __restrict__ br2, float* __restrict__ base) {
  __shared__ uint32_t keys[10];
  __shared__ float rv[256];
  __shared__ int ri[256];
  int k = blockIdx.x;    // particle
  int j = threadIdx.x;   // class
  if (j == 0) {
    // jax.random.split(key(42), 5): counts = iota(10), lower/upper halves
    uint32_t o0[5], o1[5];
    for (int q = 0; q < 5; q++)
      threefry2x32(0u, 42u, (uint32_t)q, (uint32_t)(q + 5), o0[q], o1[q]);
    for (int q = 0; q < 5; q++) { keys[q] = o0[q]; keys[q + 5] = o1[q]; }
  }
  __syncthreads();
  uint32_t kres0 = keys[0], kres1 = keys[1];

  // categorical via Gumbel argmax (uniform shape (256,256), row-major)
  const float tiny = 1.17549435e-38f;
  uint32_t bits = rbits_even(kres0, kres1, (uint32_t)(k * KP + j),
                             (uint32_t)(KP * KP));
  float g = -__logf(-__logf(jax_uniform(bits, tiny, 1.0f)));
  rv[j] = logw[j] + g;
  ri[j] = j;
  __syncthreads();
  for (int st = 128; st > 0; st >>= 1) {
    if (j < st) {
      // strict >: on ties keep the left (lower-index) entry, like jnp.argmax
      if (rv[j + st] > rv[j]) { rv[j] = rv[j + st]; ri[j] = ri[j + st]; }
    }
    __syncthreads();
  }

  if (j == 0) {
    int p = ri[0];
    uint32_t ki0 = keys[2],  ki1 = keys[3];
    uint32_t ko0 = keys[4],  ko1 = keys[5];
    uint32_t ku10 = keys[6], ku11 = keys[7];
    uint32_t ku20 = keys[8], ku21 = keys[9];

    // pair proposal
    uint32_t bi1 = rbits_even(ki0, ki1, (uint32_t)k, (uint32_t)KP);
    uint32_t bof = rbits_even(ko0, ko1, (uint32_t)k, (uint32_t)KP);
    int iv  = (int)(bi1 % 16u);
    int off = 1 + (int)(bof % 15u);
    int jv  = (iv + off) & 15;
    int i1 = min(iv, jv), i2 = max(iv, jv);

    // exponential branch lengths
    float u1 = jax_uniform(rbits_even(ku10, ku11, (uint32_t)k, (uint32_t)KP),
                           1e-6f, 1.0f);
    float u2 = jax_uniform(rbits_even(ku20, ku21, (uint32_t)k, (uint32_t)KP),
                           1e-6f, 1.0f);
    float b1v = -0.1f * __logf(u1);
    float b2v = -0.1f * __logf(u2);

    // merged leaf counts -> topology prior + log_v_minus
    int lcv[TS];
    for (int t = 0; t < TS; t++) lcv[t] = lc[p * TS + t];
    float tsum = 0.0f; int cnt = 0;
    for (int t = 0; t < TS - 1; t++) {
      int v = (t == i1) ? (lcv[i1] + lcv[i2]) : ((t == i2) ? lcv[TS - 1] : lcv[t]);
      int ti = 2 * v - 3; ti = ti < 0 ? 0 : (ti > 63 ? 63 : ti);
      if (v >= 2) tsum += -ldf[ti];
      if (v > 1) cnt++;
    }
    float lvm = __logf(fmaxf((float)cnt, 1.0f));

    // branch-length prior over all r+1 merges
    float sb = 0.0f;
    for (int r = 0; r < RM; r++) sb += b1r[p * RM + r] + b2r[p * RM + r];
    sb += b1v + b2v;
    float prior = -sb / 0.1f - 2.0f * 17.0f * __logf(0.1f);

    idxP[k] = p; idx1a[k] = i1; idx2a[k] = i2;
    br1[k] = b1v; br2[k] = b2v;
    base[k] = tsum + prior - logpi[p] + lvm - __logf(120.0f);
  }
}

// ---------------------------------------------------------------------------
// Kernel 2: sW[s][a] = site_SxC @ W_site via V_WMMA_F32_16X16X4_F32.
// W_site staged in LDS; all lane reads unconditional (mask-multiply instead
// of predicated loads -> no exec save/restore around VMEM).
// ---------------------------------------------------------------------------
__global__ __launch_bounds__(256) void k_sw(const float* __restrict__ site,
                                            const float* __restrict__ Wsite,
                                            float* __restrict__ sW) {
  __shared__ float wsh[CF * AD];
  int tid = threadIdx.x;
  if (tid < CF * AD) wsh[tid] = Wsite[tid];
  __syncthreads();

  int wave = tid >> 5, lane = tid & 31;
  int tile = blockIdx.x * 8 + wave;       // 64 tiles of 16 sites
  int s0 = tile * 16;
  int m = lane & 15, hi = lane >> 4;
  int mc = m & 3;
  float bmask = (m < AD) ? 1.0f : 0.0f;
  v8f c = {0.f, 0.f, 0.f, 0.f, 0.f, 0.f, 0.f, 0.f};
#pragma unroll
  for (int c0 = 0; c0 < CF; c0 += 4) {
    int kk = c0 + 2 * hi;
    v2f a = *(const v2f*)(site + (size_t)(s0 + m) * CF + kk);  // 8B aligned
    v2f b = { wsh[kk * AD + mc] * bmask, wsh[(kk + 1) * AD + mc] * bmask };
    c = __builtin_amdgcn_wmma_f32_16x16x4_f32(false, a, false, b, (short)0, c,
                                              false, false);
  }
  if (m < AD) {
#pragma unroll
    for (int r = 0; r < 8; r++) sW[(s0 + r + 8 * hi) * AD + m] = c[r];
  }
}

// ---------------------------------------------------------------------------
// Kernel 3: eW[k][t][a] = emb_new[k][t] @ W_stat via WMMA (merge applied).
// Merge expressed as an unconditional weighted two-row blend -> no divergent
// loads; W_stat staged in LDS.
// ---------------------------------------------------------------------------
__global__ __launch_bounds__(256) void k_ew(
    const float* __restrict__ emb, const float* __restrict__ Wstat,
    const int* __restrict__ idxP, const int* __restrict__ idx1a,
    const int* __restrict__ idx2a, float* __restrict__ eW) {
  __shared__ float wsh[DE * AD];
  int tid = threadIdx.x;
  wsh[tid] = Wstat[tid];   // 256 threads, 256 floats
  __syncthreads();

  int wave = tid >> 5, lane = tid & 31;
  int k = blockIdx.x * 8 + wave;
  int p = idxP[k], i1 = idx1a[k], i2 = idx2a[k];
  int m = lane & 15, hi = lane >> 4;
  const float* Ep = emb + (size_t)p * TS * DE;

  // per-lane source rows and blend weights (selects, not branches)
  int srcA, srcB; float wA, wB;
  if (m == 15)      { srcA = 0;      srcB = 0;  wA = 0.0f; wB = 0.0f; } // pad row
  else if (m == i1) { srcA = i1;     srcB = i2; wA = 0.5f; wB = 0.5f; } // merged
  else if (m == i2) { srcA = TS - 1; srcB = 0;  wA = 1.0f; wB = 0.0f; } // moved
  else              { srcA = m;      srcB = 0;  wA = 1.0f; wB = 0.0f; }
  const float* rowA = Ep + srcA * DE;
  const float* rowB = Ep + srcB * DE;

  int mc = m & 3;
  float bmask = (m < AD) ? 1.0f : 0.0f;
  v8f c = {0.f, 0.f, 0.f, 0.f, 0.f, 0.f, 0.f, 0.f};
  for (int c0 = 0; c0 < DE; c0 += 4) {
    int kk = c0 + 2 * hi;
    v2f ea = *(const v2f*)(rowA + kk);   // 8B aligned (kk even)
    v2f eb = *(const v2f*)(rowB + kk);
    v2f a = { ea.x * wA + eb.x * wB, ea.y * wA + eb.y * wB };
    v2f b = { wsh[kk * AD + mc] * bmask, wsh[(kk + 1) * AD + mc] * bmask };
    c = __builtin_amdgcn_wmma_f32_16x16x4_f32(false, a, false, b, (short)0, c,
                                              false, false);
  }
  if (m < AD) {
#pragma unroll
    for (int r = 0; r < 8; r++) eW[((size_t)k * TS + r + 8 * hi) * AD + m] = c[r];
  }
}

// ---------------------------------------------------------------------------
// Kernel 4: streaming likelihood pass. One block per (particle, slot).
// sW (16KB, reused by every block) is staged into LDS with
// GLOBAL_LOAD_ASYNC_TO_LDS_B128 (ASYNCcnt) and read via ds_load_b128,
// halving the per-block L2 load traffic. For t == idx1 the expm is the
// closed form exp(tQ) = e^{-t} I + (1-e^{-t}) 1*stat^T (Q^2 = -Q).
// ---------------------------------------------------------------------------
__global__ __launch_bounds__(256) void k_L(
    const float* __restrict__ logf, const float* __restrict__ sW,
    const float* __restrict__ eW, const int* __restrict__ idxP,
    const int* __restrict__ idx1a, const int* __restrict__ idx2a,
    const float* __restrict__ br1, const float* __restrict__ br2,
    float* __restrict__ L) {
  __shared__ float sWl[SS * AD];   // 16 KB
  __shared__ float red[256];
  int t = blockIdx.x;              // 0..14 (slot after merge)
  int k = blockIdx.y;              // particle
  int tid = threadIdx.x;

  // async-stage the stationary-logit site table into LDS (4 x b128 per lane)
#pragma unroll
  for (int q = 0; q < 4; q++) {
    int idx = (tid + q * 256) * AD;                       // float index, 16B
    uint32_t loff = (uint32_t)(uintptr_t)(&sWl[idx]);     // LDS byte offset
    uint64_t gaddr = (uint64_t)(uintptr_t)(sW + idx);
    asm volatile("global_load_async_to_lds_b128 %0, %1, off"
                 :: "v"(loff), "v"(gaddr) : "memory");
  }

  int p = idxP[k], i1 = idx1a[k], i2 = idx2a[k];
  float ew0 = eW[((size_t)k * TS + t) * AD + 0];
  float ew1 = eW[((size_t)k * TS + t) * AD + 1];
  float ew2 = eW[((size_t)k * TS + t) * AD + 2];
  float ew3 = eW[((size_t)k * TS + t) * AD + 3];

  asm volatile("s_wait_asynccnt 0x0" ::: "memory");
  __syncthreads();

  float acc = 0.0f;
  if (t == i1) {
    float e1 = __expf(-br1[k]), g1 = 1.0f - e1;
    float e2 = __expf(-br2[k]), g2 = 1.0f - e2;
    const float* F1 = logf + ((size_t)p * TS + i1) * SS * AD;
    const float* F2 = logf + ((size_t)p * TS + i2) * SS * AD;
    for (int s = tid; s < SS; s += 256) {
      float4 lf1 = *(const float4*)(F1 + (size_t)s * AD);
      float4 lf2 = *(const float4*)(F2 + (size_t)s * AD);
      float4 sw  = *(const float4*)(&sWl[s * AD]);
      float la0 = ew0 + sw.x, la1 = ew1 + sw.y, la2 = ew2 + sw.z, la3 = ew3 + sw.w;
      float mx = fmaxf(fmaxf(la0, la1), fmaxf(la2, la3));
      float p0 = __expf(la0 - mx), p1 = __expf(la1 - mx);
      float p2 = __expf(la2 - mx), p3 = __expf(la3 - mx);
      float psum = p0 + p1 + p2 + p3;
      float inv = 1.0f / psum;
      float f10 = __expf(lf1.x), f11 = __expf(lf1.y), f12 = __expf(lf1.z), f13 = __expf(lf1.w);
      float f20 = __expf(lf2.x), f21 = __expf(lf2.y), f22 = __expf(lf2.z), f23 = __expf(lf2.w);
      float dot1 = (p0 * f10 + p1 * f11 + p2 * f12 + p3 * f13) * inv;
      float dot2 = (p0 * f20 + p1 * f21 + p2 * f22 + p3 * f23) * inv;
      float nf0 = __logf(e1 * f10 + g1 * dot1 + 1e-30f) + __logf(e2 * f20 + g2 * dot2 + 1e-30f);
      float nf1 = __logf(e1 * f11 + g1 * dot1 + 1e-30f) + __logf(e2 * f21 + g2 * dot2 + 1e-30f);
      float nf2 = __logf(e1 * f12 + g1 * dot1 + 1e-30f) + __logf(e2 * f22 + g2 * dot2 + 1e-30f);
      float nf3 = __logf(e1 * f13 + g1 * dot1 + 1e-30f) + __logf(e2 * f23 + g2 * dot2 + 1e-30f);
      float x0 = nf0 + la0, x1 = nf1 + la1, x2 = nf2 + la2, x3 = nf3 + la3;
      float mx2 = fmaxf(fmaxf(x0, x1), fmaxf(x2, x3));
      float lse_x = mx2 + __logf(__expf(x0 - mx2) + __expf(x1 - mx2) +
                                 __expf(x2 - mx2) + __expf(x3 - mx2));
      acc += lse_x - (mx + __logf(psum));
    }
  } else {
    int su = (t == i2) ? (TS - 1) : t;   // slot idx2 holds the moved last row
    const float* F = logf + ((size_t)p * TS + su) * SS * AD;
    for (int s = tid; s < SS; s += 256) {
      if (s + 256 < SS) __builtin_prefetch(F + (size_t)(s + 256) * AD, 0, 0);
      float4 lf = *(const float4*)(F + (size_t)s * AD);
      float4 sw = *(const float4*)(&sWl[s * AD]);
      float la0 = ew0 + sw.x, la1 = ew1 + sw.y, la2 = ew2 + sw.z, la3 = ew3 + sw.w;
      float mx = fmaxf(fmaxf(la0, la1), fmaxf(la2, la3));
      float psum = __expf(la0 - mx) + __expf(la1 - mx) + __expf(la2 - mx) + __expf(la3 - mx);
      float x0 = lf.x + la0, x1 = lf.y + la1, x2 = lf.z + la2, x3 = lf.w + la3;
      float mx2 = fmaxf(fmaxf(x0, x1), fmaxf(x2, x3));
      float lse_x = mx2 + __logf(__expf(x0 - mx2) + __expf(x1 - mx2) +
                                 __expf(x2 - mx2) + __expf(x3 - mx2));
      acc += lse_x - (mx + __logf(psum));
    }
  }

  red[tid] = acc;
  __syncthreads();
  for (int st = 128; st > 0; st >>= 1) {
    if (tid < st) red[tid] += red[tid + st];
    __syncthreads();
  }
  if (tid == 0) L[(size_t)k * TS + t] = red[0];
}

// ---------------------------------------------------------------------------
// Kernel 5: final weight
// ---------------------------------------------------------------------------
__global__ __launch_bounds__(256) void k_final(const float* __restrict__ L,
                                               const float* __restrict__ base,
                                               float* __restrict__ out) {
  int k = threadIdx.x;
  float s = base[k];
#pragma unroll
  for (int t = 0; t < TS - 1; t++) s += L[(size_t)k * TS + t];
  out[k] = s;
}

// ---------------------------------------------------------------------------
extern "C" void kernel_launch(void* const* d_in, const int* in_sizes, int n_in,
                              void* d_out, int out_size, void* d_ws, size_t ws_size,
                              hipStream_t stream) {
  const float* logw  = (const float*)d_in[0];
  const float* b1r   = (const float*)d_in[1];
  const float* b2r   = (const float*)d_in[2];
  const int*   lc    = (const int*)  d_in[3];
  const float* emb   = (const float*)d_in[4];
  const float* logf  = (const float*)d_in[5];
  const float* logpi = (const float*)d_in[6];
  const float* site  = (const float*)d_in[7];
  const float* Wstat = (const float*)d_in[8];
  const float* Wsite = (const float*)d_in[9];
  const float* ldf   = (const float*)d_in[10];
  float* out = (float*)d_out;

  // workspace layout (floats): ~102 KB total
  float* wsf = (float*)d_ws;
  int*   wsi = (int*)d_ws;
  int*   idxP = wsi;               // [256]
  int*   i1a  = wsi + 256;         // [256]
  int*   i2a  = wsi + 512;         // [256]
  float* br1  = wsf + 768;         // [256]
  float* br2  = wsf + 1024;        // [256]
  float* base = wsf + 1280;        // [256]
  float* eW   = wsf + 1536;        // [256][16][4]
  float* sW   = eW + KP * TS * AD; // [1024][4]
  float* L    = sW + SS * AD;      // [256][16]

  k_setup<<<KP, 256, 0, stream>>>(logw, b1r, b2r, lc, logpi, ldf,
                                  idxP, i1a, i2a, br1, br2, base);
  k_sw<<<8, 256, 0, stream>>>(site, Wsite, sW);
  k_ew<<<32, 256, 0, stream>>>(emb, Wstat, idxP, i1a, i2a, eW);
  dim3 gL(TS - 1, KP);
  k_L<<<gL, 256, 0, stream>>>(logf, sW, eW, idxP, i1a, i2a, br1, br2, L);
  k_final<<<1, 256, 0, stream>>>(L, base, out);
}